// DeepLatent_49589692399911
// MI455X (gfx1250) — compile-verified
//
#include <hip/hip_runtime.h>

typedef __attribute__((ext_vector_type(2))) float v2f;
typedef __attribute__((ext_vector_type(8))) float v8f;

#define B_   64
#define N_   1024
#define L_   64
#define H_   128
#define CIN_ 67      // 3 + L
#define PTS  128     // points per MLP block
#define STRP 288     // H1 k-pair row stride in floats: [64 kpairs][PTS*2 + 32 pad]
                     //   lane-half row offset 288 % 64banks == 32 -> disjoint bank ranges
#define STR2 136     // H2 row stride (floats), phase-3 reads are lane-consecutive

// ---------------------------------------------------------------------------
// Kernel A: per-batch latent bias  c1[b][ch] = b1[ch] + sum_l W1[ch][3+l]*latent[b][l]
// ---------------------------------------------------------------------------
__global__ void k_c1(const float* __restrict__ W1, const float* __restrict__ b1,
                     const float* __restrict__ latent, float* __restrict__ c1) {
    const int b = blockIdx.x, ch = threadIdx.x;
    const float* w   = W1 + ch * CIN_ + 3;
    const float* lat = latent + b * L_;
    float acc = b1[ch];
#pragma unroll 8
    for (int l = 0; l < L_; ++l) acc = fmaf(w[l], lat[l], acc);
    c1[b * H_ + ch] = acc;
}

// ---------------------------------------------------------------------------
// Kernel B: fused MLP.  Grid: (N/PTS, B), 256 threads (8 waves).
//   phase1: H1 = relu(W1[:, :3] @ pc_tile + c1)   (K=3, VALU) -> LDS k-pair interleaved
//   phase2: H2 = relu(W2 @ H1 + b2)  via V_WMMA_F32_16X16X4_F32, B frags = 1x ds_load_b64
//   phase3: noise = W3 @ H2 + b3 ; pc_est = pc - noise -> global
// ---------------------------------------------------------------------------
__global__ void k_mlp(const float* __restrict__ pc, const float* __restrict__ W1,
                      const float* __restrict__ c1, const float* __restrict__ W2,
                      const float* __restrict__ b2, const float* __restrict__ W3,
                      const float* __restrict__ b3, float* __restrict__ pc_est) {
    extern __shared__ float smem[];
    float* sH1 = smem;                    // [H_/2][STRP]   (k-pair, pt, 2)
    float* sH2 = smem + (H_ / 2) * STRP;  // [H_][STR2]     (ch-major)

    const int b   = blockIdx.y;
    const int n0  = blockIdx.x * PTS;
    const int tid = threadIdx.x;
    const int pt  = tid & (PTS - 1);
    const int hlf = tid >> 7;

    // ---- phase 1: layer 1 (K collapsed to 3: latent rows are constant over points)
    const float px = pc[(b * 3 + 0) * N_ + n0 + pt];
    const float py = pc[(b * 3 + 1) * N_ + n0 + pt];
    const float pz = pc[(b * 3 + 2) * N_ + n0 + pt];
#pragma unroll 4
    for (int ch = hlf * 64; ch < hlf * 64 + 64; ch += 2) {
        const float* w0 = W1 + ch * CIN_;
        const float* w1 = w0 + CIN_;
        float h0 = c1[b * H_ + ch];
        float h1 = c1[b * H_ + ch + 1];
        h0 = fmaf(w0[0], px, h0); h0 = fmaf(w0[1], py, h0); h0 = fmaf(w0[2], pz, h0);
        h1 = fmaf(w1[0], px, h1); h1 = fmaf(w1[1], py, h1); h1 = fmaf(w1[2], pz, h1);
        v2f hp;
        hp.x = fmaxf(h0, 0.0f);
        hp.y = fmaxf(h1, 0.0f);
        *(v2f*)&sH1[(ch >> 1) * STRP + pt * 2] = hp;   // one b64 store, conflict-free
    }
    __syncthreads();

    // ---- phase 2: layer 2 via f32 WMMA. wave 'wv' owns output channels [16wv,16wv+16)
    const int wv  = tid >> 5;
    const int ln  = tid & 31;
    const int lh  = ln >> 4;     // lane half: selects K pair for A/B, M+8 for C/D
    const int l16 = ln & 15;
    const int arow = wv * 16 + l16;      // W2 row feeding the A fragment

    __builtin_prefetch(W2 + arow * H_, 0, 1);   // global_prefetch_b8

    v8f C[8];
#pragma unroll
    for (int t = 0; t < 8; ++t) {
#pragma unroll
        for (int r = 0; r < 8; ++r) C[t][r] = 0.0f;
    }

#pragma unroll 4
    for (int ks = 0; ks < 32; ++ks) {
        const int kpair = ks * 2 + lh;                  // k = {2*kpair, 2*kpair+1}
        // A fragment: lanes 0-15 K={4ks,4ks+1}, lanes 16-31 K={4ks+2,4ks+3}
        const v2f A = *(const v2f*)&W2[arow * H_ + ks * 4 + lh * 2];  // global_load_b64
        const float* brow = &sH1[kpair * STRP];
#pragma unroll
        for (int t = 0; t < 8; ++t) {
            const int col = t * 16 + l16;
            const v2f Bf = *(const v2f*)&brow[col * 2];  // single aligned ds_load_b64
            // 8 args: (neg_a, A, neg_b, B, c_mod, C, reuse_a, reuse_b)
            C[t] = __builtin_amdgcn_wmma_f32_16x16x4_f32(
                false, A, false, Bf, (short)0, C[t], false, false);
        }
    }

    // bias + relu; C layout: VGPR r holds M = r + 8*lane_half, N = l16
#pragma unroll
    for (int t = 0; t < 8; ++t) {
#pragma unroll
        for (int r = 0; r < 8; ++r) {
            const int ch = wv * 16 + lh * 8 + r;
            sH2[ch * STR2 + t * 16 + l16] = fmaxf(C[t][r] + b2[ch], 0.0f);
        }
    }
    __syncthreads();

    // ---- phase 3: layer 3 (M=3, K=128) + residual store
    if (tid < PTS) {
        float a0 = b3[0], a1 = b3[1], a2 = b3[2];
#pragma unroll 4
        for (int k = 0; k < H_; ++k) {
            const float h = sH2[k * STR2 + tid];
            a0 = fmaf(W3[0 * H_ + k], h, a0);
            a1 = fmaf(W3[1 * H_ + k], h, a1);
            a2 = fmaf(W3[2 * H_ + k], h, a2);
        }
        pc_est[(b * 3 + 0) * N_ + n0 + tid] = px - a0;
        pc_est[(b * 3 + 1) * N_ + n0 + tid] = py - a1;
        pc_est[(b * 3 + 2) * N_ + n0 + tid] = pz - a2;
    }
}

// ---------------------------------------------------------------------------
// Kernel C: per-batch chamfer partials + L2 partial. Grid: B, 256 threads.
//   partial[b] = { sum_n min_m d(gt_n, est_m), sum_m min_n d, sum (gt-est)^2 }
// ---------------------------------------------------------------------------
__global__ void k_chamfer(const float* __restrict__ pc_gt, const float* __restrict__ pc_est,
                          float* __restrict__ partial) {
    __shared__ float sGT[3 * N_];
    __shared__ float sES[3 * N_];
    __shared__ float sRed[3 * 256];
    const int b = blockIdx.x, tid = threadIdx.x;

    for (int i = tid; i < 3 * N_; i += 256) {
        sGT[i] = pc_gt[b * 3 * N_ + i];
        sES[i] = pc_est[b * 3 * N_ + i];
    }
    __syncthreads();

    float sumsq = 0.0f;
    for (int i = tid; i < 3 * N_; i += 256) {
        const float d = sGT[i] - sES[i];
        sumsq = fmaf(d, d, sumsq);
    }

    float srow = 0.0f;   // gt -> nearest est
    for (int n = tid; n < N_; n += 256) {
        const float gx = sGT[n], gy = sGT[N_ + n], gz = sGT[2 * N_ + n];
        float mn = 3.4e38f;
#pragma unroll 4
        for (int m = 0; m < N_; ++m) {
            const float dx = gx - sES[m];
            const float dy = gy - sES[N_ + m];
            const float dz = gz - sES[2 * N_ + m];
            mn = fminf(mn, fmaf(dx, dx, fmaf(dy, dy, dz * dz)));
        }
        srow += mn;
    }

    float scol = 0.0f;   // est -> nearest gt
    for (int n = tid; n < N_; n += 256) {
        const float ex = sES[n], ey = sES[N_ + n], ez = sES[2 * N_ + n];
        float mn = 3.4e38f;
#pragma unroll 4
        for (int m = 0; m < N_; ++m) {
            const float dx = ex - sGT[m];
            const float dy = ey - sGT[N_ + m];
            const float dz = ez - sGT[2 * N_ + m];
            mn = fminf(mn, fmaf(dx, dx, fmaf(dy, dy, dz * dz)));
        }
        scol += mn;
    }

    sRed[tid] = srow; sRed[256 + tid] = scol; sRed[512 + tid] = sumsq;
    __syncthreads();
    for (int s = 128; s > 0; s >>= 1) {
        if (tid < s) {
            sRed[tid]       += sRed[tid + s];
            sRed[256 + tid] += sRed[256 + tid + s];
            sRed[512 + tid] += sRed[512 + tid + s];
        }
        __syncthreads();
    }
    if (tid == 0) {
        partial[b * 3 + 0] = sRed[0];
        partial[b * 3 + 1] = sRed[256];
        partial[b * 3 + 2] = sRed[512];
    }
}

// ---------------------------------------------------------------------------
// Kernel D: final scalar reduction across batches -> d_out[0..2]
// ---------------------------------------------------------------------------
__global__ void k_final(const float* __restrict__ partial, float* __restrict__ out) {
    if (threadIdx.x == 0) {
        float sr = 0.0f, sc = 0.0f, sq = 0.0f;
        for (int b = 0; b < B_; ++b) {
            sr += partial[b * 3 + 0];
            sc += partial[b * 3 + 1];
            sq += partial[b * 3 + 2];
        }
        const float inv = 1.0f / (float)(B_ * N_);
        const float chamfer = sr * inv + sc * inv;
        const float l2 = sq / (float)(B_ * 3 * N_);
        out[0] = 0.1f * chamfer + 0.9f * l2;
        out[1] = chamfer;
        out[2] = l2;
    }
}

// ---------------------------------------------------------------------------
extern "C" void kernel_launch(void* const* d_in, const int* in_sizes, int n_in,
                              void* d_out, int out_size, void* d_ws, size_t ws_size,
                              hipStream_t stream) {
    (void)in_sizes; (void)n_in; (void)out_size; (void)ws_size;
    const float* pc     = (const float*)d_in[0];
    const float* pc_gt  = (const float*)d_in[1];
    const float* latent = (const float*)d_in[2];
    const float* W1     = (const float*)d_in[3];
    const float* b1     = (const float*)d_in[4];
    const float* W2     = (const float*)d_in[5];
    const float* b2     = (const float*)d_in[6];
    const float* W3     = (const float*)d_in[7];
    const float* b3     = (const float*)d_in[8];

    float* out     = (float*)d_out;
    float* pc_est  = out + 3;                  // [B,3,N] flat, in-place in d_out
    float* c1      = (float*)d_ws;             // B*H floats
    float* partial = c1 + B_ * H_;             // B*3 floats

    k_c1<<<B_, H_, 0, stream>>>(W1, b1, latent, c1);

    dim3 gB(N_ / PTS, B_);
    const size_t shB = (size_t)((H_ / 2) * STRP + H_ * STR2) * sizeof(float); // ~143 KB of 320 KB WGP LDS
    k_mlp<<<gB, 256, shB, stream>>>(pc, W1, c1, W2, b2, W3, b3, pc_est);

    k_chamfer<<<B_, 256, 0, stream>>>(pc_gt, pc_est, partial);
    k_final<<<1, 32, 0, stream>>>(partial, out);
}